// ConvAttention3D_80427557585213
// MI455X (gfx1250) — compile-verified
//
#include <hip/hip_runtime.h>
#include <hip/hip_bf16.h>

// ---------------------------------------------------------------------------
// ConvAttention3D for MI455X (gfx1250), fp32 end-to-end via V_WMMA_F32_16X16X4_F32
//
// Exact simplifications used (softmax shift-invariance, not approximations):
//  * g_scores is constant over the softmax axis (k,l) -> cancels exactly.
//  * attn therefore does not depend on the v-index -> output replicated over v.
// ---------------------------------------------------------------------------

typedef float v2f __attribute__((ext_vector_type(2)));
typedef float v8f __attribute__((ext_vector_type(8)));

#define WMMA_F32X4(A, B, C) \
  __builtin_amdgcn_wmma_f32_16x16x4_f32(false, (A), false, (B), (short)0, (C), false, false)

#define CIN   96
#define COUT  192
#define NH    6
#define HD    32
#define GQ    4
#define HH    28
#define WW    28
#define PQ    (GQ * HH * WW)   // 3136 query positions (g,i,j)
#define PK    (GQ * 16 * 16)   // 1024 key positions   (m,k,l)
#define SCALEF 0.07216878364870323f  // 192^-0.5
#define BN_EPS 1e-5f

// ---------------------------------------------------------------------------
// Depthwise 3x3x3 conv (stride 1, pad 1) + BatchNorm(eval)  ->  t[96][3136]
// ---------------------------------------------------------------------------
__global__ __launch_bounds__(256) void dwbn_s1_kernel(
    const float* __restrict__ x, const float* __restrict__ w,
    const float* __restrict__ ga, const float* __restrict__ be,
    const float* __restrict__ mu, const float* __restrict__ va,
    float* __restrict__ t) {
  int idx = blockIdx.x * 256 + threadIdx.x;
  if (idx >= CIN * PQ) return;
  int ch = idx / PQ, p = idx % PQ;
  int g = p / 784, r = p % 784, i = r / 28, j = r % 28;
  const float* xc = x + ch * PQ;
  const float* wc = w + ch * 27;
  float acc = 0.f;
  #pragma unroll
  for (int dz = 0; dz < 3; ++dz) {
    int z = g + dz - 1;
    if ((unsigned)z >= 4u) continue;
    #pragma unroll
    for (int dy = 0; dy < 3; ++dy) {
      int yy = i + dy - 1;
      if ((unsigned)yy >= 28u) continue;
      #pragma unroll
      for (int dx = 0; dx < 3; ++dx) {
        int xx = j + dx - 1;
        if ((unsigned)xx >= 28u) continue;
        acc += xc[(z * 28 + yy) * 28 + xx] * wc[(dz * 3 + dy) * 3 + dx];
      }
    }
  }
  float inv = ga[ch] * rsqrtf(va[ch] + BN_EPS);
  t[idx] = (acc - mu[ch]) * inv + be[ch];
}

// ---------------------------------------------------------------------------
// Depthwise 3x3x3 conv (stride (1,2,2), pad (1,3,3)) + BN  ->  t[96][1024]
// ---------------------------------------------------------------------------
__global__ __launch_bounds__(256) void dwbn_s2_kernel(
    const float* __restrict__ x, const float* __restrict__ w,
    const float* __restrict__ ga, const float* __restrict__ be,
    const float* __restrict__ mu, const float* __restrict__ va,
    float* __restrict__ t) {
  int idx = blockIdx.x * 256 + threadIdx.x;
  if (idx >= CIN * PK) return;
  int ch = idx / PK, p = idx % PK;
  int g = p / 256, r = p % 256, io = r / 16, jo = r % 16;
  const float* xc = x + ch * PQ;
  const float* wc = w + ch * 27;
  float acc = 0.f;
  #pragma unroll
  for (int dz = 0; dz < 3; ++dz) {
    int z = g + dz - 1;
    if ((unsigned)z >= 4u) continue;
    #pragma unroll
    for (int dy = 0; dy < 3; ++dy) {
      int yy = io * 2 + dy - 3;
      if ((unsigned)yy >= 28u) continue;
      #pragma unroll
      for (int dx = 0; dx < 3; ++dx) {
        int xx = jo * 2 + dx - 3;
        if ((unsigned)xx >= 28u) continue;
        acc += xc[(z * 28 + yy) * 28 + xx] * wc[(dz * 3 + dy) * 3 + dx];
      }
    }
  }
  float inv = ga[ch] * rsqrtf(va[ch] + BN_EPS);
  t[idx] = (acc - mu[ch]) * inv + be[ch];
}

// ---------------------------------------------------------------------------
// 1x1 projection:  out[192][P] = W[192][96] * t[96][P] + bias   (WMMA f32)
// One wave per 16x16 output tile; K = 96 -> 24 k-steps of 4.
// ---------------------------------------------------------------------------
__global__ __launch_bounds__(256) void proj_kernel(
    const float* __restrict__ t, const float* __restrict__ W,
    const float* __restrict__ bias, float* __restrict__ out,
    int P, int ntilesP) {
  int tile = blockIdx.x * 8 + (threadIdx.x >> 5);
  if (tile >= (COUT / 16) * ntilesP) return;   // whole-wave uniform
  int ot = tile / ntilesP, pt = tile % ntilesP;
  int lane = threadIdx.x & 31;
  int mrow = lane & 15;
  int hi = lane >> 4;
  int kb = hi * 2;
  int pcol = pt * 16 + mrow;
  v8f acc = {};
  for (int kk = 0; kk < CIN / 4; ++kk) {
    int kc = kk * 4 + kb;
    v2f a, b;
    a.x = W[(ot * 16 + mrow) * CIN + kc];
    a.y = W[(ot * 16 + mrow) * CIN + kc + 1];
    b.x = t[kc * P + pcol];
    b.y = t[(kc + 1) * P + pcol];
    acc = WMMA_F32X4(a, b, acc);
  }
  #pragma unroll
  for (int r = 0; r < 8; ++r) {
    int oc = ot * 16 + r + 8 * hi;
    out[oc * P + pt * 16 + mrow] = acc[r] + bias[oc];
  }
}

// ---------------------------------------------------------------------------
// Fused attention.  Block = (head h, 16-query ij-tile), 128 threads = 4 waves.
// Wave w owns key segment m=w (256 keys).  Per g in 0..3:
//   S[16x256] = q^T k   (WMMA, K=32)  -> scaled scores into swizzled LDS
//   row softmax over 256 keys (one m)  -> attn in LDS
//   O[16x32] += attn[16x256] x v^T[256x32]   (WMMA, K=256)
// Then reduce the 4 per-wave partial O's and fold channel = c*6 + h.
// LDS swizzle: phys(row,col) = row*1024 + (col ^ (row<<2))  -> conflict-free
// for both the C-layout stores and the A-fragment reads, within 64 KB.
// ---------------------------------------------------------------------------
__global__ __launch_bounds__(128) void attn_kernel(
    const float* __restrict__ q, const float* __restrict__ k,
    const float* __restrict__ v, float* __restrict__ outattn) {
  __shared__ float Sbuf[16 * 1024];   // 64 KB

  int wid = threadIdx.x >> 5;   // wave id == m segment
  int lane = threadIdx.x & 31;
  int h = blockIdx.x / 49;
  int ijt = blockIdx.x % 49;
  int nseg = wid * 256;
  int mrow = lane & 15;
  int hi = lane >> 4;
  int kb = hi * 2;

  const float* qh = q + h * HD * PQ;
  const float* kh = k + h * HD * PK;
  const float* vh = v + h * HD * PK;

  v8f o0 = {}, o1 = {};

  for (int g = 0; g < 4; ++g) {
    // ---- hoist Q A-fragments for all 8 k-steps
    int qcol = g * 784 + ijt * 16 + mrow;
    v2f afr[8];
    #pragma unroll
    for (int kk = 0; kk < 8; ++kk) {
      afr[kk].x = qh[(kk * 4 + kb) * PQ + qcol];
      afr[kk].y = qh[(kk * 4 + kb + 1) * PQ + qcol];
    }
    // ---- S = q^T k over this wave's 16 n-tiles
    for (int nt = 0; nt < 16; ++nt) {
      int n0 = nseg + nt * 16;
      v8f acc = {};
      #pragma unroll
      for (int kk = 0; kk < 8; ++kk) {
        v2f b;
        b.x = kh[(kk * 4 + kb) * PK + n0 + mrow];
        b.y = kh[(kk * 4 + kb + 1) * PK + n0 + mrow];
        acc = WMMA_F32X4(afr[kk], b, acc);
      }
      #pragma unroll
      for (int r = 0; r < 8; ++r) {
        int row = r + 8 * hi;
        int col = n0 + mrow;
        Sbuf[row * 1024 + (col ^ (row << 2))] = acc[r] * SCALEF;
      }
    }
    // ---- softmax per query row over this wave's 256 keys (one m)
    for (int row = 0; row < 16; ++row) {
      float vals[8];
      float mx = -3.402823466e38f;
      #pragma unroll
      for (int tt = 0; tt < 8; ++tt) {
        int col = nseg + tt * 32 + lane;
        vals[tt] = Sbuf[row * 1024 + (col ^ (row << 2))];
        mx = fmaxf(mx, vals[tt]);
      }
      #pragma unroll
      for (int off = 16; off >= 1; off >>= 1)
        mx = fmaxf(mx, __shfl_xor(mx, off, 32));
      float sum = 0.f;
      #pragma unroll
      for (int tt = 0; tt < 8; ++tt) {
        vals[tt] = __expf(vals[tt] - mx);
        sum += vals[tt];
      }
      #pragma unroll
      for (int off = 16; off >= 1; off >>= 1)
        sum += __shfl_xor(sum, off, 32);
      float inv = 1.0f / sum;
      #pragma unroll
      for (int tt = 0; tt < 8; ++tt) {
        int col = nseg + tt * 32 + lane;
        Sbuf[row * 1024 + (col ^ (row << 2))] = vals[tt] * inv;
      }
    }
    // ---- O += attn[16 x 256] * v^T[256 x 32]   (2 n-tiles over c)
    for (int kk = 0; kk < 64; ++kk) {
      int kc = nseg + kk * 4 + kb;
      v2f a, b0, b1;
      a.x = Sbuf[mrow * 1024 + ((kc) ^ (mrow << 2))];
      a.y = Sbuf[mrow * 1024 + ((kc + 1) ^ (mrow << 2))];
      b0.x = vh[mrow * PK + kc];
      b0.y = vh[mrow * PK + kc + 1];
      b1.x = vh[(16 + mrow) * PK + kc];
      b1.y = vh[(16 + mrow) * PK + kc + 1];
      o0 = WMMA_F32X4(a, b0, o0);
      o1 = WMMA_F32X4(a, b1, o1);
    }
  }

  // ---- cross-wave reduction of O partials via reused LDS
  __syncthreads();
  #pragma unroll
  for (int r = 0; r < 8; ++r) {
    int ij = r + 8 * hi;
    Sbuf[wid * 1024 + ij * 32 + mrow]      = o0[r];
    Sbuf[wid * 1024 + ij * 32 + 16 + mrow] = o1[r];
  }
  __syncthreads();
  for (int e = threadIdx.x; e < 512; e += 128) {
    float s = Sbuf[e] + Sbuf[1024 + e] + Sbuf[2048 + e] + Sbuf[3072 + e];
    int ij = e >> 5, c = e & 31;
    // fold 'b c h v i j -> b (c h) v i j' : channel = c*NH + h
    outattn[(c * NH + h) * 784 + ijt * 16 + ij] = s;
  }
}

// ---------------------------------------------------------------------------
// Final 1x1 conv:  final[o][v][ij] = po_w[32][192] * oat[192][784] + po_b
// Output replicated over v (attn is exactly v-invariant).  WMMA, K=192.
// ---------------------------------------------------------------------------
__global__ __launch_bounds__(256) void outconv_kernel(
    const float* __restrict__ oat, const float* __restrict__ Wo,
    const float* __restrict__ bo, float* __restrict__ out) {
  int tile = blockIdx.x * 8 + (threadIdx.x >> 5);
  if (tile >= 2 * 49) return;   // whole-wave uniform
  int ot = tile / 49, pt = tile % 49;
  int lane = threadIdx.x & 31;
  int mrow = lane & 15;
  int hi = lane >> 4;
  int kb = hi * 2;
  v8f acc = {};
  for (int kk = 0; kk < COUT / 4; ++kk) {
    int kc = kk * 4 + kb;
    v2f a, b;
    a.x = Wo[(ot * 16 + mrow) * COUT + kc];
    a.y = Wo[(ot * 16 + mrow) * COUT + kc + 1];
    b.x = oat[kc * 784 + pt * 16 + mrow];
    b.y = oat[(kc + 1) * 784 + pt * 16 + mrow];
    acc = WMMA_F32X4(a, b, acc);
  }
  #pragma unroll
  for (int r = 0; r < 8; ++r) {
    int o = ot * 16 + r + 8 * hi;
    float val = acc[r] + bo[o];
    int base = o * (4 * 784) + pt * 16 + mrow;
    out[base]            = val;
    out[base + 784]      = val;
    out[base + 2 * 784]  = val;
    out[base + 3 * 784]  = val;
  }
}

// ---------------------------------------------------------------------------
extern "C" void kernel_launch(void* const* d_in, const int* in_sizes, int n_in,
                              void* d_out, int out_size, void* d_ws, size_t ws_size,
                              hipStream_t stream) {
  const float* x     = (const float*)d_in[0];
  const float* dwq_w = (const float*)d_in[3];
  const float* bnq_g = (const float*)d_in[4];
  const float* bnq_b = (const float*)d_in[5];
  const float* bnq_m = (const float*)d_in[6];
  const float* bnq_v = (const float*)d_in[7];
  const float* dwk_w = (const float*)d_in[8];
  const float* bnk_g = (const float*)d_in[9];
  const float* bnk_b = (const float*)d_in[10];
  const float* bnk_m = (const float*)d_in[11];
  const float* bnk_v = (const float*)d_in[12];
  const float* dwv_w = (const float*)d_in[13];
  const float* bnv_g = (const float*)d_in[14];
  const float* bnv_b = (const float*)d_in[15];
  const float* bnv_m = (const float*)d_in[16];
  const float* bnv_v = (const float*)d_in[17];
  const float* pq_w  = (const float*)d_in[18];
  const float* pq_b  = (const float*)d_in[19];
  const float* pk_w  = (const float*)d_in[20];
  const float* pk_b  = (const float*)d_in[21];
  const float* pv_w  = (const float*)d_in[22];
  const float* pv_b  = (const float*)d_in[23];
  const float* po_w  = (const float*)d_in[24];
  const float* po_b  = (const float*)d_in[25];
  float* out = (float*)d_out;

  // workspace layout (floats)
  float* ws  = (float*)d_ws;
  float* tq  = ws;                       // 96*3136 = 301056
  float* tk  = tq + CIN * PQ;            // 96*1024 =  98304
  float* tv  = tk + CIN * PK;            //            98304
  float* qb  = tv + CIN * PK;            // 192*3136 = 602112
  float* kb  = qb + COUT * PQ;           // 192*1024 = 196608
  float* vb  = kb + COUT * PK;           //            196608
  float* oat = vb + COUT * PK;           // 192*784  = 150528

  // depthwise conv + BN
  dwbn_s1_kernel<<<(CIN * PQ + 255) / 256, 256, 0, stream>>>(
      x, dwq_w, bnq_g, bnq_b, bnq_m, bnq_v, tq);
  dwbn_s2_kernel<<<(CIN * PK + 255) / 256, 256, 0, stream>>>(
      x, dwk_w, bnk_g, bnk_b, bnk_m, bnk_v, tk);
  dwbn_s2_kernel<<<(CIN * PK + 255) / 256, 256, 0, stream>>>(
      x, dwv_w, bnv_g, bnv_b, bnv_m, bnv_v, tv);

  // 1x1 projections (WMMA)
  proj_kernel<<<(12 * (PQ / 16)) / 8, 256, 0, stream>>>(tq, pq_w, pq_b, qb, PQ, PQ / 16);
  proj_kernel<<<(12 * (PK / 16)) / 8, 256, 0, stream>>>(tk, pk_w, pk_b, kb, PK, PK / 16);
  proj_kernel<<<(12 * (PK / 16)) / 8, 256, 0, stream>>>(tv, pv_w, pv_b, vb, PK, PK / 16);

  // fused attention (WMMA QK^T + softmax + WMMA AV), 6 heads x 49 ij-tiles
  attn_kernel<<<NH * 49, 128, 0, stream>>>(qb, kb, vb, oat);

  // final 1x1 conv (WMMA), replicated over v
  outconv_kernel<<<(2 * 49 + 7) / 8, 256, 0, stream>>>(oat, po_w, po_b, out);
}